// KernelPointCosmo_59820304499243
// MI455X (gfx1250) — compile-verified
//
#include <hip/hip_runtime.h>

// ---------------------------------------------------------------------------
// KPConv-style edge conv:
//   nn[e]   = argmin_k ||hood[e] - mu[k]||^2          (K = 15)
//   oe[e,o] = sum_i features[source[e], i] * w[o, nn[e], i]   (I = O = 32)
//   out[target[e], :] += oe[e, :]                      (M = 25000)
//
// Edges are binned by nn[e] (15 distinct 32x32 weight matrices); each wave32
// processes one 16-edge tile of one bin as a [16x32]x[32x32] fp32 GEMM via
// V_WMMA_F32_16X16X4_F32 (8 K-steps per 16-wide output half; B fragments
// loaded per half to stay spill-free, both accumulators kept live), then a
// single fused scatter: one ds_bpermute pair per row + two
// global_atomic_add_f32 at base/base+64B.
// ---------------------------------------------------------------------------

typedef __attribute__((ext_vector_type(2))) float v2f;
typedef __attribute__((ext_vector_type(8))) float v8f;

#define CH 32           // I == O == 32 channels

// ---- kernel 1: nearest kernel point + histogram ---------------------------
__global__ void nn_count_kernel(const float* __restrict__ hood,
                                const float* __restrict__ mu,
                                int* __restrict__ nn,
                                int* __restrict__ counts,
                                int E, int Kpts) {
    __shared__ float smu[48];            // Kpts*3 <= 45
    int tid = threadIdx.x;
    if (tid < Kpts * 3) smu[tid] = mu[tid];
    __syncthreads();
    int e = blockIdx.x * blockDim.x + tid;
    if (e >= E) return;
    float x = hood[e * 3 + 0], y = hood[e * 3 + 1], z = hood[e * 3 + 2];
    int best = 0;
    float bd = 3.4e38f;
    for (int k = 0; k < Kpts; ++k) {
        float dx = x - smu[k * 3 + 0];
        float dy = y - smu[k * 3 + 1];
        float dz = z - smu[k * 3 + 2];
        float d = dx * dx + dy * dy + dz * dz;
        if (d < bd) { bd = d; best = k; }   // strict '<' == argmin first-match
    }
    nn[e] = best;
    atomicAdd(&counts[best], 1);
}

// ---- kernel 2: tiny serial scan (K = 15) ----------------------------------
__global__ void scan_kernel(const int* __restrict__ counts,
                            int* __restrict__ offsets,
                            int* __restrict__ tileOff,
                            int Kpts) {
    if (blockIdx.x == 0 && threadIdx.x == 0) {
        int acc = 0, tacc = 0;
        for (int k = 0; k < Kpts; ++k) {
            offsets[k] = acc;
            tileOff[k] = tacc;
            acc  += counts[k];
            tacc += (counts[k] + 15) >> 4;
        }
        offsets[Kpts] = acc;
        tileOff[Kpts] = tacc;
    }
}

// ---- kernel 3: scatter edge ids into bins ---------------------------------
__global__ void fill_bins_kernel(const int* __restrict__ nn,
                                 const int* __restrict__ offsets,
                                 int* __restrict__ cursor,
                                 int* __restrict__ bins,
                                 int E) {
    int e = blockIdx.x * blockDim.x + threadIdx.x;
    if (e >= E) return;
    int k = nn[e];
    int pos = atomicAdd(&cursor[k], 1);
    bins[offsets[k] + pos] = e;
}

// ---- kernel 4: one wave32 per 16-edge tile, fp32 WMMA GEMM + scatter ------
__global__ void kpconv_gemm_kernel(const int* __restrict__ source,
                                   const int* __restrict__ target,
                                   const float* __restrict__ features,
                                   const float* __restrict__ w,      // [O, K, I]
                                   const int* __restrict__ bins,
                                   const int* __restrict__ offsets,  // K+1
                                   const int* __restrict__ tileOff,  // K+1
                                   float* __restrict__ out,          // [M, O]
                                   int Kpts) {
    const int lane = threadIdx.x & 31;
    const int wave = blockIdx.x * (blockDim.x >> 5) + (threadIdx.x >> 5);

    const int totalTiles = tileOff[Kpts];
    if (wave >= totalTiles) return;       // wave-uniform exit

    // locate bin k such that tileOff[k] <= wave < tileOff[k+1]
    int k = 0;
    for (; k < Kpts - 1 && wave >= tileOff[k + 1]; ++k) {}
    const int t   = wave - tileOff[k];
    const int off = offsets[k];
    const int cnt = offsets[k + 1] - off;
    const int p0  = t * 16;

    const int half = lane >> 4;           // 0: K rows {0,1} / M 0..7 ; 1: K rows {2,3} / M 8..15
    const int nlo  = lane & 15;
    const int kb   = half * 2;            // K offset inside each 4-wide step

    // ---- A fragments, branch-free: clamp index (tile exists => cnt >= 1) --
    const int  pA  = p0 + nlo;
    const int  pAc = (pA < cnt) ? pA : (cnt - 1);
    const bool vA  = pA < cnt;
    const float mask = vA ? 1.0f : 0.0f;
    const int eidA = bins[off + pAc];
    const int srcA = source[eidA];
    const int tgtA = target[eidA];        // row target, broadcast later via shfl
    const float* __restrict__ frow = features + (long)srcA * CH;
    v2f a[8];
#pragma unroll
    for (int s = 0; s < 8; ++s) {
        v2f val = *(const v2f*)(frow + 4 * s + kb);   // unconditional b64 load
        a[s].x = val.x * mask;                        // folds to v_pk_mul, no exec juggling
        a[s].y = val.y * mask;
    }
    const int vAi = vA ? 1 : 0;

    const long wkstride = (long)Kpts * CH;
    const float* __restrict__ wk = w + (long)k * CH;

    // ---- two 16-wide output halves; B loaded per half (spill-free),
    //      both accumulators kept live for a single fused scatter ----------
    v8f c0 = {};
    v8f c1 = {};
    {
        v2f b[8];
        const float* __restrict__ wrow0 = wk + (long)nlo * wkstride;
#pragma unroll
        for (int s = 0; s < 8; ++s)
            b[s] = *(const v2f*)(wrow0 + 4 * s + kb);
#pragma unroll
        for (int s = 0; s < 8; ++s)
            c0 = __builtin_amdgcn_wmma_f32_16x16x4_f32(false, a[s], false, b[s],
                                                       (short)0, c0, false, false);
    }
    {
        v2f b[8];
        const float* __restrict__ wrow1 = wk + (long)(nlo + 16) * wkstride;
#pragma unroll
        for (int s = 0; s < 8; ++s)
            b[s] = *(const v2f*)(wrow1 + 4 * s + kb);
#pragma unroll
        for (int s = 0; s < 8; ++s)
            c1 = __builtin_amdgcn_wmma_f32_16x16x4_f32(false, a[s], false, b[s],
                                                       (short)0, c1, false, false);
    }

    // ---- fused scatter: VGPR r of C/D holds (m = r + 8*half, n = nlo) -----
    // Row m's target/validity live in lane m; one bpermute pair per row,
    // two atomics per row sharing one base (+64B immediate for the hi half).
#pragma unroll
    for (int r = 0; r < 8; ++r) {
        int m   = r + 8 * half;
        int tgt = __shfl(tgtA, m);
        int vm  = __shfl(vAi,  m);
        if (vm) {
            float* p = out + (long)tgt * CH + nlo;
            atomicAdd(p,      c0[r]);
            atomicAdd(p + 16, c1[r]);
        }
    }
}

// ---------------------------------------------------------------------------
extern "C" void kernel_launch(void* const* d_in, const int* in_sizes, int n_in,
                              void* d_out, int out_size, void* d_ws, size_t ws_size,
                              hipStream_t stream) {
    const int*   source   = (const int*)d_in[0];
    const int*   target   = (const int*)d_in[1];
    const float* features = (const float*)d_in[2];
    const float* hood     = (const float*)d_in[3];
    const float* mu       = (const float*)d_in[4];
    const float* w        = (const float*)d_in[5];
    float* out = (float*)d_out;

    const int E    = in_sizes[0];
    const int Kpts = in_sizes[4] / 3;     // 15

    // workspace layout (ints): nn[E] | bins[E] | counts[16] | offsets[32] | tileOff[32] | cursor[16]
    int* nn      = (int*)d_ws;
    int* bins    = nn + E;
    int* counts  = bins + E;
    int* offsets = counts + 16;
    int* tileOff = offsets + 32;
    int* cursor  = tileOff + 32;

    hipMemsetAsync(out, 0, (size_t)out_size * sizeof(float), stream);
    hipMemsetAsync(counts, 0, 96 * sizeof(int), stream);   // counts+offsets+tileOff+cursor

    nn_count_kernel<<<(E + 255) / 256, 256, 0, stream>>>(hood, mu, nn, counts, E, Kpts);
    scan_kernel<<<1, 32, 0, stream>>>(counts, offsets, tileOff, Kpts);
    fill_bins_kernel<<<(E + 255) / 256, 256, 0, stream>>>(nn, offsets, cursor, bins, E);

    // upper bound on total tiles: sum_k ceil(c_k/16) <= E/16 + Kpts
    const int maxTiles = E / 16 + Kpts + 1;
    const int wavesPerBlock = 8;          // 256 threads = 8 wave32
    const int blocks = (maxTiles + wavesPerBlock - 1) / wavesPerBlock;
    kpconv_gemm_kernel<<<blocks, 256, 0, stream>>>(source, target, features, w,
                                                   bins, offsets, tileOff, out, Kpts);
}